// Block2x2DiagProductRectangular_7954279432743
// MI455X (gfx1250) — compile-verified
//
#include <hip/hip_runtime.h>
#include <hip/hip_bf16.h>

typedef float v2f __attribute__((ext_vector_type(2)));
typedef float v8f __attribute__((ext_vector_type(8)));
typedef int   v4i __attribute__((ext_vector_type(4)));

#define BATCH      16384
#define NFEAT      1024
#define ROWS_PER_WG 16
#define ROW_STRIDE 1028   // padded LDS row stride (floats); 1028*4=4112=16*257 -> float4 aligned
#define THREADS    256

// LDS layout (float offsets)
#define LDS_X   0                    // 16 x 1028 tile
#define LDS_W1  (16 * ROW_STRIDE)    // 256:   W1B[k*16+n]  = W1[n][k]
#define LDS_W2  (LDS_W1 + 256)       // 4096:  W2B[n0*256 + k*16 + n] = W2[n0][n][k]
#define LDS_W3  (LDS_W2 + 4096)      // 4096:  W3T[r*16 + i2*4 + k2]  = W3[r][i2][k2]
#define LDS_FLOATS (LDS_W3 + 4096)   // 24896 floats = 99,584 B

#define AS1 __attribute__((address_space(1)))
#define AS3 __attribute__((address_space(3)))

#if defined(__has_builtin)
#if __has_builtin(__builtin_amdgcn_global_load_async_to_lds_b128)
#define HAVE_ASYNC_LDS 1
#endif
#endif

__device__ __forceinline__ void wait_asynccnt0() {
#if defined(__has_builtin)
#if __has_builtin(__builtin_amdgcn_s_wait_asynccnt)
    __builtin_amdgcn_s_wait_asynccnt(0);
    return;
#else
    asm volatile("s_wait_asynccnt 0" ::: "memory");
    return;
#endif
#else
    asm volatile("s_wait_asynccnt 0" ::: "memory");
#endif
}

__global__ __launch_bounds__(THREADS)
void butterfly1024_wmma_kernel(
    const float* __restrict__ x,
    const float* __restrict__ a512, const float* __restrict__ a256,
    const float* __restrict__ a128, const float* __restrict__ a64,
    const float* __restrict__ a32,  const float* __restrict__ a16,
    const float* __restrict__ a8,   const float* __restrict__ a4,
    const float* __restrict__ a2,   const float* __restrict__ a1,
    const float* __restrict__ bias,
    float* __restrict__ out)
{
    extern __shared__ float lds[];
    const int tid  = threadIdx.x;
    const int lane = tid & 31;
    const int wave = tid >> 5;
    const int row0 = blockIdx.x * ROWS_PER_WG;

    // ------ load 16x1024 x-tile into LDS: async DMA (overlaps table build) --
    {
        const float* xg = x + (size_t)row0 * NFEAT;
        #pragma unroll
        for (int i = 0; i < 16; ++i) {
            int f4 = i * THREADS + tid;       // 4096 float4 total
            int b  = f4 >> 8;                 // 256 float4 per row
            int c4 = f4 & 255;
#if defined(HAVE_ASYNC_LDS)
            __builtin_amdgcn_global_load_async_to_lds_b128(
                (AS1 v4i*)(xg + b * NFEAT + c4 * 4),
                (AS3 v4i*)&lds[LDS_X + b * ROW_STRIDE + c4 * 4],
                0, 0);
#else
            float4 v = *(const float4*)(xg + b * NFEAT + c4 * 4);
            *(float4*)&lds[LDS_X + b * ROW_STRIDE + c4 * 4] = v;
#endif
        }
    }

    // ---------------- build W2 tables: thread t = n0*16 + c (column c) -----
    {
        int n0 = tid >> 4, c = tid & 15;
        float v[16];
        #pragma unroll
        for (int t = 0; t < 16; ++t) v[t] = (t == c) ? 1.f : 0.f;
        const float* As[4] = {a16, a32, a64, a128};
        #pragma unroll
        for (int s = 0; s < 4; ++s) {
            int hh = 1 << s;          // butterfly distance in n1 units
            int half = 16 << s;       // abcd last-dim size
            const float* A = As[s];
            float nv[16];
            for (int t = 0; t < 16; ++t) {
                int i_ = (t >> s) & 1;
                int j  = (t & (hh - 1)) * 16 + n0;
                float c0 = A[(i_ * 2 + 0) * half + j];
                float c1 = A[(i_ * 2 + 1) * half + j];
                nv[t] = c0 * v[t & ~hh] + c1 * v[t | hh];
            }
            for (int t = 0; t < 16; ++t) v[t] = nv[t];
        }
        for (int t = 0; t < 16; ++t)
            lds[LDS_W2 + n0 * 256 + c * 16 + t] = v[t];
    }
    // ---------------- W1 table (single 16x16): threads 0..15 ----------------
    if (tid < 16) {
        int c = tid;
        float v[16];
        #pragma unroll
        for (int t = 0; t < 16; ++t) v[t] = (t == c) ? 1.f : 0.f;
        const float* As[4] = {a1, a2, a4, a8};
        #pragma unroll
        for (int s = 0; s < 4; ++s) {
            int hh = 1 << s;          // == half for this group
            const float* A = As[s];
            float nv[16];
            for (int t = 0; t < 16; ++t) {
                int i_ = (t >> s) & 1;
                int j  = t & (hh - 1);
                float c0 = A[(i_ * 2 + 0) * hh + j];
                float c1 = A[(i_ * 2 + 1) * hh + j];
                nv[t] = c0 * v[t & ~hh] + c1 * v[t | hh];
            }
            for (int t = 0; t < 16; ++t) v[t] = nv[t];
        }
        for (int t = 0; t < 16; ++t)
            lds[LDS_W1 + c * 16 + t] = v[t];
    }
    // ---------------- W3 tables (256 x 4x4): thread r = n1*16+n0 ------------
    {
        int r = tid;
        float m[4][4];
        #pragma unroll
        for (int i = 0; i < 4; ++i)
            #pragma unroll
            for (int k = 0; k < 4; ++k) m[i][k] = (i == k) ? 1.f : 0.f;
        const float* As[2] = {a256, a512};
        #pragma unroll
        for (int s = 0; s < 2; ++s) {
            int hh = 1 << s;
            int half = 256 << s;
            const float* A = As[s];
            float nm[4][4];
            for (int t = 0; t < 4; ++t) {
                int i_ = (t >> s) & 1;
                int j  = (t & (hh - 1)) * 256 + r;
                float c0 = A[(i_ * 2 + 0) * half + j];
                float c1 = A[(i_ * 2 + 1) * half + j];
                for (int cc = 0; cc < 4; ++cc)
                    nm[t][cc] = c0 * m[t & ~hh][cc] + c1 * m[t | hh][cc];
            }
            for (int t = 0; t < 4; ++t)
                for (int cc = 0; cc < 4; ++cc) m[t][cc] = nm[t][cc];
        }
        for (int i = 0; i < 4; ++i)
            for (int k = 0; k < 4; ++k)
                lds[LDS_W3 + r * 16 + i * 4 + k] = m[i][k];
    }

#if defined(HAVE_ASYNC_LDS)
    wait_asynccnt0();     // our wave's async tile stores to LDS complete
#endif
    __syncthreads();

    // ---------------- Phase 1: W1 on contiguous 16-blocks (WMMA) ------------
    // D(16b x 16n) = A(16b x 16k) * B(16k x 16n), B[k][n] = W1[n][k]
    {
        const int mrow = lane & 15;
        const int koff = (lane < 16) ? 0 : 2;
        const int mbase = (lane < 16) ? 0 : 8;
        v2f bf[4];                      // W1 B-fragments: block-invariant
        #pragma unroll
        for (int q = 0; q < 4; ++q) {
            int kb = q * 4 + koff;
            bf[q].x = lds[LDS_W1 + kb * 16 + mrow];
            bf[q].y = lds[LDS_W1 + (kb + 1) * 16 + mrow];
        }
        #pragma unroll 2
        for (int it = 0; it < 8; ++it) {
            int base = ((it << 3) | wave) * 16;     // constant trip count
            v8f acc = {};
            #pragma unroll
            for (int q = 0; q < 4; ++q) {
                int kb = q * 4 + koff;
                v2f a;
                a.x = lds[LDS_X + mrow * ROW_STRIDE + base + kb];
                a.y = lds[LDS_X + mrow * ROW_STRIDE + base + kb + 1];
                acc = __builtin_amdgcn_wmma_f32_16x16x4_f32(
                        false, a, false, bf[q], (short)0, acc, false, false);
            }
            #pragma unroll
            for (int v = 0; v < 8; ++v)
                lds[LDS_X + (mbase + v) * ROW_STRIDE + base + mrow] = acc[v];
        }
    }
    __syncthreads();

    // ---------------- Phase 2: W2[n0] on stride-16 groups (WMMA) ------------
    {
        const int nlan = lane & 15;
        const int koff = (lane < 16) ? 0 : 2;
        const int mbase = (lane < 16) ? 0 : 8;
        #pragma unroll
        for (int h = 0; h < 2; ++h) {
            const int n0 = wave + h * 8;            // 2 distinct n0 per wave
            v2f bf[4];                              // W2[n0] B-fragments
            #pragma unroll
            for (int q = 0; q < 4; ++q) {
                int kb = q * 4 + koff;
                bf[q].x = lds[LDS_W2 + n0 * 256 + kb * 16 + nlan];
                bf[q].y = lds[LDS_W2 + n0 * 256 + (kb + 1) * 16 + nlan];
            }
            #pragma unroll 2
            for (int n2 = 0; n2 < 4; ++n2) {
                int base = n2 * 256 + n0;
                v8f acc = {};
                #pragma unroll
                for (int q = 0; q < 4; ++q) {
                    int kb = q * 4 + koff;
                    v2f a;
                    a.x = lds[LDS_X + nlan * ROW_STRIDE + base + kb * 16];
                    a.y = lds[LDS_X + nlan * ROW_STRIDE + base + (kb + 1) * 16];
                    acc = __builtin_amdgcn_wmma_f32_16x16x4_f32(
                            false, a, false, bf[q], (short)0, acc, false, false);
                }
                #pragma unroll
                for (int v = 0; v < 8; ++v)
                    lds[LDS_X + (mbase + v) * ROW_STRIDE + base + nlan * 16] = acc[v];
            }
        }
    }
    __syncthreads();

    // ---------------- Phase 3: 4x4 W3 + bias, coalesced store ---------------
    {
        int r = tid;                       // = n1*16 + n0, fixed per thread
        float w3[16];
        #pragma unroll
        for (int i = 0; i < 16; ++i) w3[i] = lds[LDS_W3 + r * 16 + i];
        float bs[4];
        #pragma unroll
        for (int i2 = 0; i2 < 4; ++i2) bs[i2] = bias[i2 * 256 + r];
        for (int b = 0; b < 16; ++b) {
            float xv[4];
            #pragma unroll
            for (int k2 = 0; k2 < 4; ++k2)
                xv[k2] = lds[LDS_X + b * ROW_STRIDE + k2 * 256 + r];
            float* og = out + (size_t)(row0 + b) * NFEAT;
            #pragma unroll
            for (int i2 = 0; i2 < 4; ++i2) {
                float acc = bs[i2];
                #pragma unroll
                for (int k2 = 0; k2 < 4; ++k2)
                    acc += w3[i2 * 4 + k2] * xv[k2];
                og[i2 * 256 + r] = acc;
            }
        }
    }
}

extern "C" void kernel_launch(void* const* d_in, const int* in_sizes, int n_in,
                              void* d_out, int out_size, void* d_ws, size_t ws_size,
                              hipStream_t stream) {
    (void)in_sizes; (void)out_size; (void)d_ws; (void)ws_size;

    const float* x = (const float*)d_in[0];
    const float* ab[10]; // ab[i] = abcd_list[i], half = 1024 >> (i+1)
    const float* bias;

    if (n_in >= 12) {
        // dict order: x, abcd_list[0..9] (halves 512..1), bias
        for (int i = 0; i < 10; ++i) ab[i] = (const float*)d_in[1 + i];
        bias = (const float*)d_in[11];
    } else {
        // fallback: abcd_list concatenated flat in d_in[1]
        const float* base = (const float*)d_in[1];
        size_t off = 0;
        for (int i = 0; i < 10; ++i) { ab[i] = base + off; off += 2u * (1024u >> i); }
        bias = (const float*)d_in[2];
    }

    dim3 grid(BATCH / ROWS_PER_WG);     // 1024 workgroups
    dim3 block(THREADS);
    size_t shmem = LDS_FLOATS * sizeof(float);  // ~99.6 KB dynamic LDS

    butterfly1024_wmma_kernel<<<grid, block, shmem, stream>>>(
        x,
        ab[0], ab[1], ab[2], ab[3], ab[4],
        ab[5], ab[6], ab[7], ab[8], ab[9],
        bias, (float*)d_out);
}